// FeatureANI_1657857376257
// MI455X (gfx1250) — compile-verified
//
#include <hip/hip_runtime.h>
#include <math.h>

typedef float v2f __attribute__((ext_vector_type(2)));
typedef float v8f __attribute__((ext_vector_type(8)));

#define NBATCH 16
#define NATOM  64
#define NRAD   16
#define NANG   32
#define RCR_F  5.2f
#define RCA_F  3.5f
#define PI_F   3.14159265358979323846f
#define GPITCH 65   // 64 + 1 pad: diagonal access conflict-free

__global__ __launch_bounds__(256) void aev_wmma_kernel(
    const float* __restrict__ coords,   // (B, N, 3)
    const int*   __restrict__ atype,    // (B, N)
    const float* __restrict__ EtaR,     // (NT, NR)
    const float* __restrict__ ShfR,     // (NT, NR)
    const float* __restrict__ Zeta,     // (NT, NA)
    const float* __restrict__ ShfZ,     // (NT, NA)
    const float* __restrict__ EtaA,     // (NT, NA)
    const float* __restrict__ ShfA,     // (NT, NA)
    float* __restrict__ out)            // (B, N, NR+NA)
{
    __shared__ float cx[NATOM], cy[NATOM], cz[NATOM];
    __shared__ float gram[NATOM * GPITCH];
    __shared__ float dd[NATOM], fR[NATOM], fA[NATOM];
    __shared__ float etaR_s[NRAD], shfR_s[NRAD];
    __shared__ float zeta_s[NANG], csA[NANG], snA[NANG];
    __shared__ float etaA_s[NANG], shfz_s[NANG], p2z[NANG];
    __shared__ float g2s[NRAD], g3s[NANG];
    __shared__ float rn2_s, rn3_s;

    const int tid = threadIdx.x;
    const int b   = blockIdx.x >> 6;   // batch
    const int i   = blockIdx.x & 63;   // center atom

    // ---- stage coordinates of this batch into LDS ----
    if (tid < NATOM) {
        const float* cp = coords + ((size_t)b * NATOM + tid) * 3;
        cx[tid] = cp[0]; cy[tid] = cp[1]; cz[tid] = cp[2];
    }
    // ---- per-center-atom parameters (type of atom i) ----
    const int ti = atype[b * NATOM + i];
    if (tid < NRAD) {
        etaR_s[tid] = EtaR[ti * NRAD + tid];
        shfR_s[tid] = ShfR[ti * NRAD + tid];
        g2s[tid] = 0.f;
    }
    if (tid < NANG) {
        float z = Zeta[ti * NANG + tid];
        zeta_s[tid] = z;
        p2z[tid]    = __powf(2.f, 1.f - z);
        float sa    = ShfA[ti * NANG + tid];
        csA[tid]    = __cosf(sa);
        snA[tid]    = __sinf(sa);
        etaA_s[tid] = EtaA[ti * NANG + tid];
        shfz_s[tid] = ShfZ[ti * NANG + tid];
        g3s[tid] = 0.f;
    }
    __syncthreads();

    // ---- 64x64 Gram matrix via V_WMMA_F32_16X16X4_F32 (K=3 padded to 4) ----
    // A 16x4 f32 layout: lanes 0-15 -> (K0,K1) = (x,y); lanes 16-31 -> (K2,K3) = (z,0)
    // B 4x16 f32 layout is symmetric (rows K striped across lanes per VGPR).
    const int  lane = tid & 31;
    const int  wv   = tid >> 5;      // 8 waves
    const int  lm   = lane & 15;
    const bool hiL  = lane >= 16;
    for (int t = wv; t < 16; t += 8) {          // 16 tiles, 2 per wave, uniform
        const int mi = t >> 2, nj = t & 3;
        const int mrow = mi * 16 + lm;
        const int ncol = nj * 16 + lm;
        v2f a, bv;
        a.x  = hiL ? cz[mrow] : cx[mrow];
        a.y  = hiL ? 0.f      : cy[mrow];
        bv.x = hiL ? cz[ncol] : cx[ncol];
        bv.y = hiL ? 0.f      : cy[ncol];
        v8f c = {0.f, 0.f, 0.f, 0.f, 0.f, 0.f, 0.f, 0.f};
        c = __builtin_amdgcn_wmma_f32_16x16x4_f32(
                /*neg_a=*/false, a, /*neg_b=*/false, bv,
                /*c_mod=*/(short)0, c, /*reuse_a=*/false, /*reuse_b=*/false);
        const int Mbase = mi * 16 + (hiL ? 8 : 0);
        const int Nc    = nj * 16 + lm;
        #pragma unroll
        for (int r = 0; r < 8; ++r)
            gram[(Mbase + r) * GPITCH + Nc] = c[r];
    }
    __syncthreads();

    // ---- distances & cutoffs from Gram ----
    const float gii = gram[i * GPITCH + i];
    if (tid < NATOM) {
        const int j = tid;
        float gjj = gram[j * GPITCH + j];
        float gij = gram[i * GPITCH + j];
        float dv  = sqrtf(fmaxf(gii + gjj - 2.f * gij, 0.f));
        float fr  = (dv < RCR_F) ? 0.5f * (__cosf(dv * (PI_F / RCR_F)) + 1.f) : 0.f;
        float fa  = (dv < RCA_F) ? 0.5f * (__cosf(dv * (PI_F / RCA_F)) + 1.f) : 0.f;
        if (j == i) { dv = 0.f; fr = 0.f; fa = 0.f; }   // exclude self
        dd[j] = dv; fR[j] = fr; fA[j] = fa;
    }
    __syncthreads();

    // ---- radial G2: 64 neighbors x 16 features across all 256 threads ----
    for (int idx = tid; idx < NATOM * NRAD; idx += 256) {
        const int j = idx >> 4, r = idx & 15;
        const float f = fR[j];
        if (f > 0.f) {
            const float x = dd[j] - shfR_s[r];
            atomicAdd(&g2s[r], __expf(-etaR_s[r] * x * x) * f);
        }
    }

    // ---- angular G3: pairs j1<j2, register accumulators ----
    float acc[NANG];
    #pragma unroll
    for (int a2 = 0; a2 < NANG; ++a2) acc[a2] = 0.f;

    for (int flat = tid; flat < NATOM * NATOM; flat += 256) {
        const int j1 = flat >> 6, j2 = flat & 63;
        if (j2 <= j1) continue;
        const float ff = fA[j1] * fA[j2];
        if (ff <= 0.f) continue;                 // also skips self-pairs (fA[i]=0)
        const float d12 = dd[j1], d13 = dd[j2];
        const float dot = gram[j1 * GPITCH + j2]
                        - gram[i  * GPITCH + j1]
                        - gram[i  * GPITCH + j2] + gii;
        float ca = dot / (d12 * d13);
        ca = fminf(1.f, fmaxf(-1.f, ca));
        const float sa   = sqrtf(fmaxf(0.f, 1.f - ca * ca));  // sin(theta) >= 0
        const float davg = 0.5f * (d12 + d13);
        #pragma unroll
        for (int a2 = 0; a2 < NANG; ++a2) {
            // cos(theta - ShfA) = cos(theta)cos(ShfA) + sin(theta)sin(ShfA)
            const float cosshift = ca * csA[a2] + sa * snA[a2];
            const float base = fmaxf(0.f, 1.f + cosshift);
            const float pw   = __powf(base, zeta_s[a2]);
            const float x    = davg - shfz_s[a2];
            const float g    = __expf(-etaA_s[a2] * x * x);
            acc[a2] += p2z[a2] * pw * g * ff;
        }
    }

    // ---- wave32 tree reduce, then one LDS atomic per wave per feature ----
    #pragma unroll
    for (int a2 = 0; a2 < NANG; ++a2) {
        float v = acc[a2];
        for (int m = 16; m >= 1; m >>= 1) v += __shfl_xor(v, m, 32);
        if (lane == 0) atomicAdd(&g3s[a2], v);
    }
    __syncthreads();

    // ---- normalization ----
    if (tid == 0) {
        float s2 = 0.f, s3 = 0.f;
        for (int r = 0; r < NRAD; ++r)  s2 += g2s[r] * g2s[r];
        for (int a2 = 0; a2 < NANG; ++a2) s3 += g3s[a2] * g3s[a2];
        const float n2 = sqrtf(s2);
        const float n3 = sqrtf(s3);
        rn2_s = 1.f / n2;                        // reference has no zero guard on g2
        rn3_s = (n3 == 0.f) ? 1.f : 1.f / n3;    // reference guards g3
    }
    __syncthreads();

    if (tid < NRAD + NANG) {
        const float v = (tid < NRAD) ? g2s[tid] * rn2_s
                                     : g3s[tid - NRAD] * rn3_s;
        out[((size_t)(b * NATOM + i)) * (NRAD + NANG) + tid] = v;
    }
}

extern "C" void kernel_launch(void* const* d_in, const int* in_sizes, int n_in,
                              void* d_out, int out_size, void* d_ws, size_t ws_size,
                              hipStream_t stream) {
    (void)in_sizes; (void)n_in; (void)d_ws; (void)ws_size; (void)out_size;
    const float* coords = (const float*)d_in[0];
    const int*   atype  = (const int*)  d_in[1];
    const float* EtaR   = (const float*)d_in[2];
    const float* ShfR   = (const float*)d_in[3];
    const float* Zeta   = (const float*)d_in[4];
    const float* ShfZ   = (const float*)d_in[5];
    const float* EtaA   = (const float*)d_in[6];
    const float* ShfA   = (const float*)d_in[7];
    float* out = (float*)d_out;

    dim3 grid(NBATCH * NATOM);   // one workgroup per (batch, atom)
    dim3 block(256);             // 8 wave32 waves
    aev_wmma_kernel<<<grid, block, 0, stream>>>(coords, atype, EtaR, ShfR,
                                                Zeta, ShfZ, EtaA, ShfA, out);
}